// ConfidenceModel_32753420599498
// MI455X (gfx1250) — compile-verified
//
#include <hip/hip_runtime.h>
#include <math.h>

typedef float v2f __attribute__((ext_vector_type(2)));
typedef float v4f __attribute__((ext_vector_type(4)));
typedef float v8f __attribute__((ext_vector_type(8)));
typedef int   v4i __attribute__((ext_vector_type(4)));

#define BATCH 4096
#define NCLS  1000
#define DIM   512
#define NCOL  8000   // 1000 classes * 8 concepts

// ---- CDNA5 async memory->LDS path (gated; falls back to plain copy) ----
#if defined(__AMDGCN__) && __has_builtin(__builtin_amdgcn_global_load_async_to_lds_b128)
#define ASYNC_COPY 1
#else
#define ASYNC_COPY 0
#endif

#if ASYNC_COPY
typedef __attribute__((address_space(1))) v4i* gptr_v4i;
typedef __attribute__((address_space(3))) v4i* lptr_v4i;
#endif

__device__ __forceinline__ void copy16_to_lds(const float* g, float* l) {
#if ASYNC_COPY
  __builtin_amdgcn_global_load_async_to_lds_b128(
      (gptr_v4i)g, (lptr_v4i)l, 0, 0);
#else
  *(v4f*)l = *(const v4f*)g;
#endif
}

__device__ __forceinline__ void wait_async_copies() {
#if ASYNC_COPY
#if __has_builtin(__builtin_amdgcn_s_wait_asynccnt)
  __builtin_amdgcn_s_wait_asynccnt(0);
#else
  asm volatile("s_wait_asynccnt 0x0" ::: "memory");
#endif
#endif
}

// ---------------- Kernel 1: inverse L2 norms of image rows ----------------
__global__ __launch_bounds__(256) void inv_norm_kernel(const float* __restrict__ img,
                                                       float* __restrict__ invn) {
  const int wave = threadIdx.x >> 5;
  const int lane = threadIdx.x & 31;
  const int row  = blockIdx.x * 8 + wave;
  const float* p = img + (size_t)row * DIM + lane * 16;
  float s = 0.f;
#pragma unroll
  for (int i = 0; i < 4; ++i) {
    v4f v = *(const v4f*)(p + i * 4);
    s += v.x * v.x + v.y * v.y + v.z * v.z + v.w * v.w;
  }
#pragma unroll
  for (int m = 16; m >= 1; m >>= 1) s += __shfl_xor(s, m, 32);
  if (lane == 0) invn[row] = 1.0f / sqrtf(s);
}

// ---------------- Kernel 2: sims GEMM via V_WMMA_F32_16X16X4_F32 ----------
// C[4096,8000] = diag(invn) * A[4096,512] * B[8000,512]^T
// 256 threads = 8 waves (2x4), wave tile 64x64, block tile 128x256.
// Double-buffered LDS, async global->LDS DMA, one barrier per K-chunk.
#define BM  128
#define BN  256
#define BK  32
#define LDA 36                          // padded LDS row stride (conflict-free)
#define BUF_FLTS (BM * LDA + BN * LDA)  // 13824 floats = 55296 B per buffer

__global__ __launch_bounds__(256, 1) void sim_gemm_kernel(
    const float* __restrict__ A,      // image encodings [4096,512]
    const float* __restrict__ Bm,     // description encodings flat [8000,512]
    const float* __restrict__ invn,   // [4096]
    float* __restrict__ C)            // sims [4096,8000]
{
  __shared__ __align__(16) float lds[2 * BUF_FLTS];   // 110592 bytes

  const int t    = threadIdx.x;
  const int lane = t & 31;
  const int wave = t >> 5;
  const int lid  = lane & 15;
  const int hi   = lane >> 4;

  const int bm = blockIdx.y * BM;
  const int bn = blockIdx.x * BN;
  const int wm = (wave >> 2) * 64;
  const int wn = (wave & 3)  * 64;

  v8f acc[4][4];
#pragma unroll
  for (int i = 0; i < 4; ++i)
#pragma unroll
    for (int j = 0; j < 4; ++j)
#pragma unroll
      for (int e = 0; e < 8; ++e) acc[i][j][e] = 0.f;

  // issue one K-chunk's A(128x32) + B(256x32) tiles into LDS buffer `buf`
  auto issue_chunk = [&](int k0, int buf) {
    float* bufA = lds + buf * BUF_FLTS;
    float* bufB = bufA + BM * LDA;
#pragma unroll
    for (int j = 0; j < 4; ++j) {               // A: 128 rows x 32 cols
      int lin = j * 256 + t;
      int r   = lin >> 3;
      int c4  = (lin & 7) * 4;
      copy16_to_lds(A + (size_t)(bm + r) * DIM + k0 + c4, bufA + r * LDA + c4);
    }
#pragma unroll
    for (int j = 0; j < 8; ++j) {               // B: 256 rows x 32 cols
      int lin = j * 256 + t;
      int r   = lin >> 3;
      int c4  = (lin & 7) * 4;
      int gr  = bn + r;
      if (gr < NCOL)                            // edge rows stay stale: their
        copy16_to_lds(Bm + (size_t)gr * DIM + k0 + c4, bufB + r * LDA + c4);
    }                                           // columns are never stored
  };

  const int NCHUNK = DIM / BK;                  // 16
  issue_chunk(0, 0);

  for (int ck = 0; ck < NCHUNK; ++ck) {
    wait_async_copies();                        // this wave's DMAs landed
    __syncthreads();                            // everyone's DMAs landed; prev
                                                // buffer free for overwrite
    if (ck + 1 < NCHUNK) issue_chunk((ck + 1) * BK, (ck + 1) & 1);

    const float* bufA = lds + (ck & 1) * BUF_FLTS;
    const float* bufB = bufA + BM * LDA;
#pragma unroll
    for (int kk = 0; kk < BK; kk += 4) {
      v2f af[4], bf[4];
#pragma unroll
      for (int tm = 0; tm < 4; ++tm)
        af[tm] = *(const v2f*)(bufA + (wm + tm * 16 + lid) * LDA + kk + 2 * hi);
#pragma unroll
      for (int tn = 0; tn < 4; ++tn)
        bf[tn] = *(const v2f*)(bufB + (wn + tn * 16 + lid) * LDA + kk + 2 * hi);
#pragma unroll
      for (int tm = 0; tm < 4; ++tm)
#pragma unroll
        for (int tn = 0; tn < 4; ++tn)
          acc[tm][tn] = __builtin_amdgcn_wmma_f32_16x16x4_f32(
              false, af[tm], false, bf[tn], (short)0, acc[tm][tn], false, false);
    }
  }

  // epilogue: scale rows by invn and store (VGPR i -> row i + 8*hi, col lid)
#pragma unroll
  for (int tm = 0; tm < 4; ++tm) {
    const int rbase = bm + wm + tm * 16 + hi * 8;
    float sc[8];
#pragma unroll
    for (int i = 0; i < 8; ++i) sc[i] = invn[rbase + i];
#pragma unroll
    for (int tn = 0; tn < 4; ++tn) {
      int col = bn + wn + tn * 16 + lid;
      if (col < NCOL) {
#pragma unroll
        for (int i = 0; i < 8; ++i)
          C[(size_t)(rbase + i) * NCOL + col] = acc[tm][tn][i] * sc[i];
      }
    }
  }
}

// ---------------- Kernel 3: top-8, votes, argmax (one block per row) ------
__global__ __launch_bounds__(256, 1) void topk_vote_kernel(
    const float* __restrict__ sims,
    float* __restrict__ preds,
    float* __restrict__ conf)
{
  __shared__ float row[NCOL];
  __shared__ float votes[NCLS];
  __shared__ float rv[256];
  __shared__ int   ri[256];

  const int t = threadIdx.x;
  const int b = blockIdx.x;
  const float* src = sims + (size_t)b * NCOL;

  for (int j = t; j < NCOL; j += 256) row[j] = src[j];
  for (int j = t; j < NCLS; j += 256) votes[j] = 0.f;
  __syncthreads();

  float wsum = 0.f;
#pragma unroll
  for (int r = 1; r <= 8; ++r) wsum += log1pf((float)r);   // ln(9!)

  for (int pass = 0; pass < 8; ++pass) {
    float bv = -INFINITY; int bi = 0x7fffffff;
    for (int j = t; j < NCOL; j += 256) {
      float v = row[j];
      if (v > bv || (v == bv && j < bi)) { bv = v; bi = j; }
    }
    rv[t] = bv; ri[t] = bi;
    __syncthreads();
    for (int s = 128; s > 0; s >>= 1) {
      if (t < s) {
        float ov = rv[t + s]; int oi = ri[t + s];
        if (ov > rv[t] || (ov == rv[t] && oi < ri[t])) { rv[t] = ov; ri[t] = oi; }
      }
      __syncthreads();
    }
    if (t == 0) {
      int idx = ri[0];
      row[idx] = -INFINITY;                                 // mask winner
      votes[idx >> 3] += log1pf((float)(8 - pass)) / wsum;  // rank weight
    }
    __syncthreads();
  }

  // argmax over votes, first-index tie-break (matches jnp.argmax)
  float bv = -INFINITY; int bi = 0x7fffffff;
  for (int j = t; j < NCLS; j += 256) {
    float v = votes[j];
    if (v > bv || (v == bv && j < bi)) { bv = v; bi = j; }
  }
  rv[t] = bv; ri[t] = bi;
  __syncthreads();
  for (int s = 128; s > 0; s >>= 1) {
    if (t < s) {
      float ov = rv[t + s]; int oi = ri[t + s];
      if (ov > rv[t] || (ov == rv[t] && oi < ri[t])) { rv[t] = ov; ri[t] = oi; }
    }
    __syncthreads();
  }
  if (t == 0) { preds[b] = (float)ri[0]; conf[b] = rv[0]; }
}

// ---------------- launcher ----------------
extern "C" void kernel_launch(void* const* d_in, const int* in_sizes, int n_in,
                              void* d_out, int out_size, void* d_ws, size_t ws_size,
                              hipStream_t stream) {
  const float* img  = (const float*)d_in[0];   // [4096,512]
  const float* desc = (const float*)d_in[1];   // [1000,8,512] -> flat [8000,512]

  float* preds = (float*)d_out;                // [4096]
  float* conf  = preds + BATCH;                // [4096]
  float* sims  = conf + BATCH;                 // [4096,8000]

  float* invn = (float*)d_ws;                  // 16 KB scratch

  inv_norm_kernel<<<dim3(BATCH / 8), 256, 0, stream>>>(img, invn);
  sim_gemm_kernel<<<dim3((NCOL + BN - 1) / BN, BATCH / BM), 256, 0, stream>>>(
      img, desc, invn, sims);
  topk_vote_kernel<<<dim3(BATCH), 256, 0, stream>>>(sims, preds, conf);
}